// LearnedPositionalEmbedding_79680233275485
// MI455X (gfx1250) — compile-verified
//
#include <hip/hip_runtime.h>
#include <hip/hip_bf16.h>

// D = A x B with V_WMMA_F32_16X16X4_F32 (wave32):
//   A: 16x4 f32  -> v2f per lane (lane m=0..15 holds M=m; lane>>4 selects K pair {0,1}/{2,3})
//   B: 4x16 f32  -> v2f per lane (lane m holds N=m; lane>>4 selects K pair)
//   C/D: 16x16 f32 -> v8f (VGPR v: row = v + 8*(lane>>4), col = lane&15)
typedef __attribute__((ext_vector_type(2))) float v2f;
typedef __attribute__((ext_vector_type(8))) float v8f;

#define NUM_HEADS 16
#define HEAD_DIM  64
#define LSEQ      1024
#define MAXPOS    1024
#define TBLW      (2 * MAXPOS + 1)   // 2049
#define WAVES_PER_WG 4

__global__ __launch_bounds__(32 * WAVES_PER_WG)
void relpos_wmma_kernel(const float* __restrict__ q,    // [B,1,H,L,D]
                        const float* __restrict__ tbl,  // [H,D,TBLW]
                        float* __restrict__ out)        // [B,1,H,L,L]
{
    const int nIB  = LSEQ / 16;
    const int blk  = blockIdx.x;
    const int iblk = blk % nIB;          // 16-row block of i
    const int bh   = blk / nIB;          // b*H + h
    const int i0   = iblk * 16;
    const int h    = bh % NUM_HEADS;

    const int lane  = threadIdx.x & 31;
    const int wave  = threadIdx.x >> 5;
    const int m     = lane & 15;         // M-row / N-col / table-col lane index
    const int khalf = lane >> 4;         // selects K pair within a 4-slice

    const float* __restrict__ Q = q   + (size_t)bh * LSEQ * HEAD_DIM + (size_t)i0 * HEAD_DIM;
    const float* __restrict__ T = tbl + (size_t)h  * HEAD_DIM * TBLW;
    float* __restrict__ O       = out + (size_t)bh * LSEQ * LSEQ;

    // ---- Load A operand once: Q[i0:i0+16, 0:64] into 16 K-slices of v2f ----
    // Slice kc covers K = 4*kc..4*kc+3; this lane supplies K = 4*kc + 2*khalf + {0,1}.
    v2f a[16];
#pragma unroll
    for (int kc = 0; kc < 16; ++kc) {
        const float* p = Q + (size_t)m * HEAD_DIM + 4 * kc + 2 * khalf;
        a[kc].x = p[0];
        a[kc].y = p[1];
    }

    // Diagonals needed for this i-block: s = j - i in [-(i0+15), 1023-i0]
    const int smin    = -(i0 + 15);
    const int smax    = (LSEQ - 1) - i0;
    const int s0_base = smin & ~15;                   // floor to multiple of 16
    const int ntiles  = (smax - s0_base) / 16 + 1;    // <= 66

    // Each wave processes pairs of adjacent 16-wide diagonal tiles (2 accumulators for ILP).
    for (int t0 = wave * 2; t0 < ntiles; t0 += WAVES_PER_WG * 2) {
        const int  s0    = s0_base + t0 * 16;
        const bool have2 = (t0 + 1) < ntiles;

        // Table column bases (clamped for safety at edge tiles; stores are masked anyway).
        int col0 = s0 + MAXPOS + m;
        int col1 = col0 + 16;
        col0 = col0 < 0 ? 0 : (col0 > TBLW - 1 ? TBLW - 1 : col0);
        col1 = col1 < 0 ? 0 : (col1 > TBLW - 1 ? TBLW - 1 : col1);

        v8f acc0 = {0.f, 0.f, 0.f, 0.f, 0.f, 0.f, 0.f, 0.f};
        v8f acc1 = {0.f, 0.f, 0.f, 0.f, 0.f, 0.f, 0.f, 0.f};

#pragma unroll
        for (int kc = 0; kc < 16; ++kc) {
            const int krow = 4 * kc + 2 * khalf;
            const float* Tk = T + (size_t)krow * TBLW;

            v2f b0;
            b0.x = Tk[col0];
            b0.y = Tk[TBLW + col0];
            acc0 = __builtin_amdgcn_wmma_f32_16x16x4_f32(
                false, a[kc], false, b0, (short)0, acc0, false, false);

            v2f b1;
            b1.x = Tk[col1];
            b1.y = Tk[TBLW + col1];
            acc1 = __builtin_amdgcn_wmma_f32_16x16x4_f32(
                false, a[kc], false, b1, (short)0, acc1, false, false);
        }

        // ---- Scatter-store the skewed tiles: D[r,n] -> out[i0+r, i0+r+s0+n] ----
#pragma unroll
        for (int v = 0; v < 8; ++v) {
            const int i = i0 + v + 8 * khalf;
            const int j = i + s0 + m;
            float* row = O + (size_t)i * LSEQ;
            if ((unsigned)j < (unsigned)LSEQ)
                row[j] = acc0[v];
            const int j2 = j + 16;
            if (have2 && (unsigned)j2 < (unsigned)LSEQ)
                row[j2] = acc1[v];
        }
    }
}

extern "C" void kernel_launch(void* const* d_in, const int* in_sizes, int n_in,
                              void* d_out, int out_size, void* d_ws, size_t ws_size,
                              hipStream_t stream) {
    (void)in_sizes; (void)n_in; (void)out_size; (void)d_ws; (void)ws_size;
    const float* q   = (const float*)d_in[0];   // [4,1,16,1024,64] f32
    const float* tbl = (const float*)d_in[1];   // [16,64,2049]     f32
    // d_in[2] is the scalar `length` (==1024), baked in at compile time.
    float* out = (float*)d_out;                 // [4,1,16,1024,1024] f32

    const int nblocks = 4 * NUM_HEADS * (LSEQ / 16);   // 4096
    relpos_wmma_kernel<<<nblocks, 32 * WAVES_PER_WG, 0, stream>>>(q, tbl, out);
}